// LogSparseAttention_76690936038081
// MI455X (gfx1250) — compile-verified
//
#include <hip/hip_runtime.h>
#include <math.h>

typedef __attribute__((ext_vector_type(2))) float v2f;
typedef __attribute__((ext_vector_type(8))) float v8f;

#define WAVES 4

constexpr int B_ = 2, L_ = 2048, S_ = 2048, H_ = 8, E_ = 64, D_ = 64;
constexpr float SCALE  = 0.125f;     // 1/sqrt(64)
constexpr float MASKED = -1.25e8f;   // SCALE * (-1e9), matches reference exactly

// mask(i,j) for win_len = sub_len = 2048, log_l = 11:
//   i < 22  : full causal (j <= i)
//   i >= 22 : d = i-j in [0,10]  OR  d-10 is a power of two <= 1024
__device__ __forceinline__ bool log_mask(int i, int j) {
    int d = i - j;
    if (d < 0) return false;
    if (i < 22) return true;
    if (d <= 10) return true;
    int e = d - 10;
    return (e <= 1024) && ((e & (e - 1)) == 0);
}

// 16x16 tile (i0,j0) contains any active (i,j)?
__device__ __forceinline__ bool tile_active(int i0, int j0) {
    int delta = i0 - j0;               // d range over tile pair: [delta-15, delta+15]
    if (delta < -15) return false;     // strictly above causal band
    if (i0 < 32) return true;          // tiles containing rows < 22: causal
    if (delta <= 25) return true;      // overlaps d in [0,10]
#pragma unroll
    for (int p = 0; p < 11; ++p) {
        int dv = 10 + (1 << p);        // {11,12,14,18,26,42,74,138,266,522,1034}
        if (dv >= delta - 15 && dv <= delta + 15) return true;
    }
    return false;
}

__global__ __launch_bounds__(WAVES * 32)
void logsparse_attn_kernel(const float* __restrict__ Q,
                           const float* __restrict__ K,
                           const float* __restrict__ V,
                           float* __restrict__ Out) {
    __shared__ __align__(16) float plds[WAVES][16][18];  // padded: 72B row stride

    const int lane = threadIdx.x & 31;
    const int wave = threadIdx.x >> 5;
    const int g    = lane >> 4;      // half-wave group (0/1)
    const int ln   = lane & 15;

    const int gid = blockIdx.x * WAVES + wave;  // 0 .. B*H*(L/16)-1 = 2047
    const int qt  = gid & 127;                  // L/16 = 128 query tiles
    const int h   = (gid >> 7) & 7;             // H = 8
    const int b   = gid >> 10;
    const int i0  = qt * 16;

    // ---- Q fragments: f32 WMMA A-layout. lane holds row (i0+ln), k = 4*kc + 2*g + v
    const float* qrow = Q + ((size_t)(b * L_ + i0 + ln) * H_ + h) * E_ + 2 * g;
    v2f qf[16];
#pragma unroll
    for (int kc = 0; kc < 16; ++kc) {
        float2 t = *(const float2*)(qrow + 4 * kc);
        v2f a; a[0] = t.x; a[1] = t.y;
        qf[kc] = a;
    }

    v8f oacc[4];
#pragma unroll
    for (int nt = 0; nt < 4; ++nt) oacc[nt] = (v8f){};
    float rowmax[8], rowsum[8];
#pragma unroll
    for (int r = 0; r < 8; ++r) { rowmax[r] = -INFINITY; rowsum[r] = 0.0f; }

    const int jt_end = qt;  // causal: j0 <= i0
    for (int jt = 0; jt <= jt_end; ++jt) {
        const int j0 = jt * 16;
        if (!tile_active(i0, j0)) continue;

        // ---- S = Q * K^T for this 16x16 tile (16 chained K=4 WMMAs)
        // B = K^T: lane holds key row (j0+ln), element k = 4*kc + 2*g + v
        const float* krow = K + ((size_t)(b * S_ + j0 + ln) * H_ + h) * E_ + 2 * g;
        v8f s = (v8f){};
#pragma unroll
        for (int kc = 0; kc < 16; ++kc) {
            float2 t = *(const float2*)(krow + 4 * kc);
            v2f kf; kf[0] = t.x; kf[1] = t.y;
            s = __builtin_amdgcn_wmma_f32_16x16x4_f32(false, qf[kc], false, kf,
                                                      (short)0, s, false, false);
        }

        // ---- mask + online softmax. C-layout: s[r] is row m = r + 8*g, col j0+ln
#pragma unroll
        for (int r = 0; r < 8; ++r) {
            const int i = i0 + r + 8 * g;
            const int j = j0 + ln;
            float logit = log_mask(i, j) ? SCALE * s[r] : MASKED;

            float tmax = logit;
#pragma unroll
            for (int off = 8; off >= 1; off >>= 1)
                tmax = fmaxf(tmax, __shfl_xor(tmax, off, 32));   // within 16-lane group

            float newmax = fmaxf(rowmax[r], tmax);
            float corr   = __expf(rowmax[r] - newmax);
            float p      = __expf(logit - newmax);

            float tsum = p;
#pragma unroll
            for (int off = 8; off >= 1; off >>= 1)
                tsum += __shfl_xor(tsum, off, 32);

            rowsum[r] = rowsum[r] * corr + tsum;
            rowmax[r] = newmax;
#pragma unroll
            for (int nt = 0; nt < 4; ++nt) oacc[nt][r] *= corr;

            plds[wave][r + 8 * g][ln] = p;   // stage P for C->A layout transpose
        }
        __builtin_amdgcn_wave_barrier();     // order DS store -> DS load (same wave)

        // ---- read P back in A-layout: lane holds row ln, k = 4*kc + 2*g + v
        v2f pa[4];
#pragma unroll
        for (int kc = 0; kc < 4; ++kc) {
            float2 t = *(const float2*)(&plds[wave][ln][4 * kc + 2 * g]);
            v2f a; a[0] = t.x; a[1] = t.y;
            pa[kc] = a;
        }
        __builtin_amdgcn_wave_barrier();     // protect LDS before next iteration

        // ---- O += P * V : 4 output tiles (D=64), 4 K=4 chunks each
        const float* vbase = V + ((size_t)(b * S_ + j0) * H_ + h) * D_;
#pragma unroll
        for (int nt = 0; nt < 4; ++nt) {
#pragma unroll
            for (int kc = 0; kc < 4; ++kc) {
                const int kr = 4 * kc + 2 * g;   // V row within tile for v=0
                v2f vf;
                vf[0] = vbase[(size_t)(kr + 0) * (H_ * D_) + nt * 16 + ln];
                vf[1] = vbase[(size_t)(kr + 1) * (H_ * D_) + nt * 16 + ln];
                oacc[nt] = __builtin_amdgcn_wmma_f32_16x16x4_f32(false, pa[kc], false, vf,
                                                                 (short)0, oacc[nt], false, false);
            }
        }
    }

    // ---- normalize and store: Out[b][i][h][d], row m = r + 8*g, col = nt*16 + ln
#pragma unroll
    for (int r = 0; r < 8; ++r) {
        const int i = i0 + r + 8 * g;
        const float inv = 1.0f / rowsum[r];
        float* orow = Out + ((size_t)(b * L_ + i) * H_ + h) * D_;
        orow[ 0 + ln] = oacc[0][r] * inv;
        orow[16 + ln] = oacc[1][r] * inv;
        orow[32 + ln] = oacc[2][r] * inv;
        orow[48 + ln] = oacc[3][r] * inv;
    }
}

extern "C" void kernel_launch(void* const* d_in, const int* in_sizes, int n_in,
                              void* d_out, int out_size, void* d_ws, size_t ws_size,
                              hipStream_t stream) {
    (void)in_sizes; (void)n_in; (void)out_size; (void)d_ws; (void)ws_size;
    const float* Q = (const float*)d_in[0];
    const float* K = (const float*)d_in[1];
    const float* V = (const float*)d_in[2];
    float* Out = (float*)d_out;

    const int total_waves = B_ * H_ * (L_ / 16);     // 2048
    dim3 grid(total_waves / WAVES);                  // 512 blocks
    dim3 block(WAVES * 32);                          // 4 wave32 per block
    hipLaunchKernelGGL(logsparse_attn_kernel, grid, block, 0, stream, Q, K, V, Out);
}